// Model_61125974557077
// MI455X (gfx1250) — compile-verified
//
#include <hip/hip_runtime.h>
#include <math.h>

#define BB   4
#define LL   512
#define DD   512
#define HH   8
#define HDIM 64
#define TD   8
#define TW   64
#define DFF  2048
#define MROWS (BB*LL)          // 2048 token rows
#define LNEPS 1e-5f

typedef __attribute__((ext_vector_type(2))) float v2f;
typedef __attribute__((ext_vector_type(8))) float v8f;

// ---------------------------------------------------------------------------
// fp32 WMMA GEMM: C[z] = A[z] @ B[z] (+bias) (+addmat) (+relu)
// One wave (32 threads) per 16x64 output tile: 4 independent accumulator
// chains (hides WMMA RAW latency), A fragment loaded once and reused 4x.
// Uses V_WMMA_F32_16X16X4_F32 (full fp32 precision).
// Requirements: M % 16 == 0, N % 64 == 0, K % 16 == 0 (true for all calls).
// flags: bit0 = relu, bit1 = B is transposed (C = A @ B^T)
// ---------------------------------------------------------------------------
__global__ __launch_bounds__(32) void gemm_wmma_f32(
    const float* __restrict__ A, const float* __restrict__ Bm,
    float* __restrict__ C,
    int K, int lda, int ldb, int ldc,
    long long sA, long long sB, long long sC,
    const float* __restrict__ bias,     // [N] or null
    const float* __restrict__ addmat,   // same geometry as C, or null
    int flags)
{
    const int lane = threadIdx.x & 31;
    const int l15  = lane & 15;
    const int kh   = lane >> 4;             // 0/1: K pair {0,1} vs {2,3}
    const int col0 = blockIdx.x * 64;
    const int row0 = blockIdx.y * 16;
    const long long zb = blockIdx.z;

    const float* Ab = A  + zb * sA;
    const float* Bb = Bm + zb * sB;
    float*       Cb = C  + zb * sC;

    v8f acc[4];
    #pragma unroll
    for (int t = 0; t < 4; ++t) acc[t] = (v8f){0.f,0.f,0.f,0.f,0.f,0.f,0.f,0.f};

    const float* arow = Ab + (long long)(row0 + l15) * lda;

    if (flags & 2) {
        // C = A @ B^T : B'[k][n] = Bm[n][k]; lane n reads row (col0+16t+n).
        const float* br[4];
        #pragma unroll
        for (int t = 0; t < 4; ++t)
            br[t] = Bb + (long long)(col0 + 16 * t + l15) * ldb;
        #pragma unroll 4
        for (int k = 0; k < K; k += 4) {
            const int kk = k + 2 * kh;
            v2f a;
            a.x = arow[kk]; a.y = arow[kk + 1];
            v2f b[4];
            #pragma unroll
            for (int t = 0; t < 4; ++t) { b[t].x = br[t][kk]; b[t].y = br[t][kk + 1]; }
            #pragma unroll
            for (int t = 0; t < 4; ++t)
                acc[t] = __builtin_amdgcn_wmma_f32_16x16x4_f32(
                    false, a, false, b[t], (short)0, acc[t], false, false);
        }
    } else {
        // lane n reads column (col0+16t+n), rows kk, kk+1.
        const float* bc[4];
        #pragma unroll
        for (int t = 0; t < 4; ++t) bc[t] = Bb + col0 + 16 * t + l15;
        #pragma unroll 4
        for (int k = 0; k < K; k += 4) {
            const int kk = k + 2 * kh;
            const long long r0 = (long long)kk * ldb;
            const long long r1 = r0 + ldb;
            v2f a;
            a.x = arow[kk]; a.y = arow[kk + 1];
            v2f b[4];
            #pragma unroll
            for (int t = 0; t < 4; ++t) { b[t].x = bc[t][r0]; b[t].y = bc[t][r1]; }
            #pragma unroll
            for (int t = 0; t < 4; ++t)
                acc[t] = __builtin_amdgcn_wmma_f32_16x16x4_f32(
                    false, a, false, b[t], (short)0, acc[t], false, false);
        }
    }

    // Epilogue. C/D layout: VGPR r -> row (r + 8*kh), lane l15 -> col.
    #pragma unroll
    for (int t = 0; t < 4; ++t) {
        const int colw = col0 + 16 * t + l15;
        const float bv = bias ? bias[colw] : 0.f;
        #pragma unroll
        for (int r = 0; r < 8; ++r) {
            const int roww = row0 + r + 8 * kh;
            const long long ci = (long long)roww * ldc + colw;
            float v = acc[t][r] + bv;
            if (addmat) v += addmat[zb * sC + ci];
            if (flags & 1) v = fmaxf(v, 0.f);
            Cb[ci] = v;
        }
    }
}

// ---------------------------------------------------------------------------
// Per-token MLP on t_emb: 8 -> 64 -> 64 -> 512 (relu, relu, linear).
// One block of 64 threads per row.
// ---------------------------------------------------------------------------
__global__ __launch_bounds__(TW) void tmlp_kernel(
    const float* __restrict__ T,    // [MROWS, 8]
    const float* __restrict__ w0, const float* __restrict__ b0,
    const float* __restrict__ w1, const float* __restrict__ b1,
    const float* __restrict__ w2, const float* __restrict__ b2,
    float* __restrict__ out)        // [MROWS, 512]
{
    __shared__ float t[TD];
    __shared__ float z0[TW];
    __shared__ float z1[TW];
    const int row = blockIdx.x;
    const int j = threadIdx.x;

    if (j < TD) t[j] = T[(long long)row * TD + j];
    __syncthreads();

    float s = b0[j];
    #pragma unroll
    for (int k = 0; k < TD; ++k) s += t[k] * w0[k * TW + j];
    z0[j] = fmaxf(s, 0.f);
    __syncthreads();

    s = b1[j];
    for (int k = 0; k < TW; ++k) s += z0[k] * w1[k * TW + j];
    z1[j] = fmaxf(s, 0.f);
    __syncthreads();

    for (int c = j; c < DD; c += TW) {
        float o = b2[c];
        for (int k = 0; k < TW; ++k) o += z1[k] * w2[k * DD + c];
        out[(long long)row * DD + c] = o;
    }
}

// ---------------------------------------------------------------------------
// Pairwise time-delta MLP: delta(b,i,j)[8] -> 64 -> 64 -> 8 heads.
// One block of 64 threads per (b,i,j). Output written as [B,H,L,L].
// ---------------------------------------------------------------------------
__global__ __launch_bounds__(TW) void amlp_kernel(
    const float* __restrict__ T,    // [B, L, 8]
    const float* __restrict__ w0, const float* __restrict__ b0,
    const float* __restrict__ w1, const float* __restrict__ b1,
    const float* __restrict__ w2,   // [64, 8]
    const float* __restrict__ b2,   // [8]
    float* __restrict__ Abias)      // [B, H, L, L]
{
    __shared__ float t[TD];
    __shared__ float z0[TW];
    __shared__ float z1[TW];
    const int blk = blockIdx.x;
    const int b   = blk / (LL * LL);
    const int rem = blk % (LL * LL);
    const int i   = rem / LL;
    const int jj  = rem % LL;
    const int j = threadIdx.x;

    if (j < TD)
        t[j] = T[((long long)b * LL + i) * TD + j] - T[((long long)b * LL + jj) * TD + j];
    __syncthreads();

    float s = b0[j];
    #pragma unroll
    for (int k = 0; k < TD; ++k) s += t[k] * w0[k * TW + j];
    z0[j] = fmaxf(s, 0.f);
    __syncthreads();

    s = b1[j];
    for (int k = 0; k < TW; ++k) s += z0[k] * w1[k * TW + j];
    z1[j] = fmaxf(s, 0.f);
    __syncthreads();

    if (j < HH) {
        float o = b2[j];
        for (int k = 0; k < TW; ++k) o += z1[k] * w2[k * HH + j];
        Abias[((((long long)b * HH + j) * LL) + i) * LL + jj] = o;
    }
}

// ---------------------------------------------------------------------------
// Permutes: [B,L,H*HD] -> [B,H,L,HD] and back
// ---------------------------------------------------------------------------
__global__ void heads_kernel(const float* __restrict__ X, float* __restrict__ Out)
{
    long long idx = (long long)blockIdx.x * blockDim.x + threadIdx.x;
    if (idx >= (long long)BB * LL * DD) return;
    const int d = (int)(idx % HDIM);
    long long r = idx / HDIM;
    const int l = (int)(r % LL); r /= LL;
    const int h = (int)(r % HH);
    const int b = (int)(r / HH);
    Out[idx] = X[((long long)b * LL + l) * DD + h * HDIM + d];
}

__global__ void unheads_kernel(const float* __restrict__ X, float* __restrict__ Out)
{
    long long idx = (long long)blockIdx.x * blockDim.x + threadIdx.x;
    if (idx >= (long long)BB * LL * DD) return;
    const int c = (int)(idx % DD);
    long long r = idx / DD;
    const int l = (int)(r % LL);
    const int b = (int)(r / LL);
    const int h = c / HDIM;
    const int d = c % HDIM;
    Out[idx] = X[(((long long)b * HH + h) * LL + l) * HDIM + d];
}

// ---------------------------------------------------------------------------
// Row softmax over L=512 of scores * Abias * (1/sqrt(HD)), in place.
// One block of 128 threads per row; each thread owns 4 columns.
// ---------------------------------------------------------------------------
__global__ __launch_bounds__(128) void softmax_kernel(
    float* __restrict__ S, const float* __restrict__ Ab)
{
    __shared__ float red[128];
    const long long row = blockIdx.x;
    float* s = S + row * LL;
    const float* ab = Ab + row * LL;
    const float scale = 0.125f;   // 1/sqrt(64)
    const int tid = threadIdx.x;

    float v[4];
    float m = -3.4e38f;
    #pragma unroll
    for (int t = 0; t < 4; ++t) {
        const int j = tid + t * 128;
        const float x = s[j] * ab[j] * scale;
        v[t] = x;
        m = fmaxf(m, x);
    }
    red[tid] = m; __syncthreads();
    for (int st = 64; st > 0; st >>= 1) {
        if (tid < st) red[tid] = fmaxf(red[tid], red[tid + st]);
        __syncthreads();
    }
    m = red[0]; __syncthreads();

    float sum = 0.f;
    #pragma unroll
    for (int t = 0; t < 4; ++t) { v[t] = expf(v[t] - m); sum += v[t]; }
    red[tid] = sum; __syncthreads();
    for (int st = 64; st > 0; st >>= 1) {
        if (tid < st) red[tid] += red[tid + st];
        __syncthreads();
    }
    const float inv = 1.f / red[0];
    #pragma unroll
    for (int t = 0; t < 4; ++t) s[tid + t * 128] = v[t] * inv;
}

// ---------------------------------------------------------------------------
// LayerNorm(X + Y) over D=512 with gain/bias. 256 threads, 2 elems each.
// ---------------------------------------------------------------------------
__global__ __launch_bounds__(256) void ln_kernel(
    const float* __restrict__ X, const float* __restrict__ Y,
    const float* __restrict__ g, const float* __restrict__ bb,
    float* __restrict__ Out)
{
    __shared__ float red[256];
    const long long row = blockIdx.x;
    const int tid = threadIdx.x;
    const float* xr = X + row * DD;
    const float* yr = Y + row * DD;
    const float v0 = xr[tid]       + yr[tid];
    const float v1 = xr[tid + 256] + yr[tid + 256];

    red[tid] = v0 + v1; __syncthreads();
    for (int st = 128; st > 0; st >>= 1) {
        if (tid < st) red[tid] += red[tid + st];
        __syncthreads();
    }
    const float mu = red[0] * (1.f / DD); __syncthreads();

    const float d0 = v0 - mu, d1 = v1 - mu;
    red[tid] = d0 * d0 + d1 * d1; __syncthreads();
    for (int st = 128; st > 0; st >>= 1) {
        if (tid < st) red[tid] += red[tid + st];
        __syncthreads();
    }
    const float rstd = rsqrtf(red[0] * (1.f / DD) + LNEPS);

    Out[row * DD + tid]       = d0 * rstd * g[tid]       + bb[tid];
    Out[row * DD + tid + 256] = d1 * rstd * g[tid + 256] + bb[tid + 256];
}

// ---------------------------------------------------------------------------
extern "C" void kernel_launch(void* const* d_in, const int* in_sizes, int n_in,
                              void* d_out, int out_size, void* d_ws, size_t ws_size,
                              hipStream_t stream) {
    (void)in_sizes; (void)out_size; (void)ws_size;
    const float* x     = (const float*)d_in[0];
    const float* t_emb = (const float*)d_in[1];
    int nLayers = (n_in - 2) / 37;
    if (nLayers < 1) nLayers = 1;

    // Workspace carve (fp32): 6 token buffers + 2 score-sized buffers ~92 MB.
    float* w = (float*)d_ws;
    size_t off = 0;
    auto carve = [&](size_t n) { float* r = w + off; off += n; return r; };
    const size_t MD = (size_t)MROWS * DD;          // 1M elems
    const size_t SS = (size_t)BB * HH * LL * LL;   // 8.4M elems
    float* tq = carve(MD); float* tk = carve(MD); float* tv = carve(MD);
    float* gq = carve(MD); float* gk = carve(MD); float* gv = carve(MD);
    float* abias  = carve(SS);
    float* scores = carve(SS);

    const int tot = BB * LL * DD;
    const float* xin = x;
    int p = 2;
    for (int layer = 0; layer < nLayers; ++layer) {
        const float* WQ = (const float*)d_in[p + 0];
        const float* WK = (const float*)d_in[p + 1];
        const float* WV = (const float*)d_in[p + 2];
        const float* WO = (const float*)d_in[p + 3];
        const float* bO = (const float*)d_in[p + 4];
        const float* mq[6]; for (int i = 0; i < 6; ++i) mq[i] = (const float*)d_in[p + 5  + i];
        const float* mk[6]; for (int i = 0; i < 6; ++i) mk[i] = (const float*)d_in[p + 11 + i];
        const float* mv[6]; for (int i = 0; i < 6; ++i) mv[i] = (const float*)d_in[p + 17 + i];
        const float* ma[6]; for (int i = 0; i < 6; ++i) ma[i] = (const float*)d_in[p + 23 + i];
        const float* ln1g = (const float*)d_in[p + 29];
        const float* ln1b = (const float*)d_in[p + 30];
        const float* ln2g = (const float*)d_in[p + 31];
        const float* ln2b = (const float*)d_in[p + 32];
        const float* Wc1  = (const float*)d_in[p + 33];
        const float* bc1  = (const float*)d_in[p + 34];
        const float* Wc2  = (const float*)d_in[p + 35];
        const float* bc2  = (const float*)d_in[p + 36];
        p += 37;

        // t-emb MLPs -> tq/tk/tv ; pairwise A-MLP -> abias [B,H,L,L]
        tmlp_kernel<<<MROWS, TW, 0, stream>>>(t_emb, mq[0],mq[1],mq[2],mq[3],mq[4],mq[5], tq);
        tmlp_kernel<<<MROWS, TW, 0, stream>>>(t_emb, mk[0],mk[1],mk[2],mk[3],mk[4],mk[5], tk);
        tmlp_kernel<<<MROWS, TW, 0, stream>>>(t_emb, mv[0],mv[1],mv[2],mv[3],mv[4],mv[5], tv);
        amlp_kernel<<<BB * LL * LL, TW, 0, stream>>>(t_emb, ma[0],ma[1],ma[2],ma[3],ma[4],ma[5], abias);

        // QKV projections (fused +tMLP): gq/gk/gv = xin@W* + t*
        gemm_wmma_f32<<<dim3(DD/64, MROWS/16, 1), 32, 0, stream>>>(
            xin, WQ, gq, DD, DD, DD, DD, 0, 0, 0, nullptr, tq, 0);
        gemm_wmma_f32<<<dim3(DD/64, MROWS/16, 1), 32, 0, stream>>>(
            xin, WK, gk, DD, DD, DD, DD, 0, 0, 0, nullptr, tk, 0);
        gemm_wmma_f32<<<dim3(DD/64, MROWS/16, 1), 32, 0, stream>>>(
            xin, WV, gv, DD, DD, DD, DD, 0, 0, 0, nullptr, tv, 0);

        // split heads: gq/gk/gv -> tq/tk/tv as [B,H,L,HD]
        heads_kernel<<<(tot + 255) / 256, 256, 0, stream>>>(gq, tq);
        heads_kernel<<<(tot + 255) / 256, 256, 0, stream>>>(gk, tk);
        heads_kernel<<<(tot + 255) / 256, 256, 0, stream>>>(gv, tv);

        // scores = Q @ K^T  (batched over B*H)
        gemm_wmma_f32<<<dim3(LL/64, LL/16, BB*HH), 32, 0, stream>>>(
            tq, tk, scores, HDIM, HDIM, HDIM, LL,
            (long long)LL * HDIM, (long long)LL * HDIM, (long long)LL * LL,
            nullptr, nullptr, 2);

        // softmax((scores * abias) / sqrt(HD)) in place
        softmax_kernel<<<BB * HH * LL, 128, 0, stream>>>(scores, abias);

        // attnout = attn @ V -> gq as [B,H,L,HD]
        gemm_wmma_f32<<<dim3(HDIM/64, LL/16, BB*HH), 32, 0, stream>>>(
            scores, tv, gq, LL, LL, HDIM, HDIM,
            (long long)LL * LL, (long long)LL * HDIM, (long long)LL * HDIM,
            nullptr, nullptr, 0);

        // merge heads -> gk [B,L,D]; output projection -> gv
        unheads_kernel<<<(tot + 255) / 256, 256, 0, stream>>>(gq, gk);
        gemm_wmma_f32<<<dim3(DD/64, MROWS/16, 1), 32, 0, stream>>>(
            gk, WO, gv, DD, DD, DD, DD, 0, 0, 0, bO, nullptr, 0);

        // x2 = LN(xin + attn_proj) -> tq
        ln_kernel<<<MROWS, 256, 0, stream>>>(xin, gv, ln1g, ln1b, tq);

        // FFN: scores buffer reused as [2048, 2048] hidden
        gemm_wmma_f32<<<dim3(DFF/64, MROWS/16, 1), 32, 0, stream>>>(
            tq, Wc1, scores, DD, DD, DFF, DFF, 0, 0, 0, bc1, nullptr, 1);
        gemm_wmma_f32<<<dim3(DD/64, MROWS/16, 1), 32, 0, stream>>>(
            scores, Wc2, gq, DFF, DFF, DD, DD, 0, 0, 0, bc2, nullptr, 0);

        // out = LN(x2 + ffn) -> d_out
        ln_kernel<<<MROWS, 256, 0, stream>>>(tq, gq, ln2g, ln2b, (float*)d_out);

        xin = (const float*)d_out;
    }
}